// RNN2Classifier_43181601194181
// MI455X (gfx1250) — compile-verified
//
#include <hip/hip_runtime.h>

typedef __attribute__((ext_vector_type(2))) float v2f;
typedef __attribute__((ext_vector_type(8))) float v8f;

// tanh(x) = 1 - 2/(exp(2x)+1); exp2 under/overflow gives exact -1/+1 saturation.
__device__ __forceinline__ float fast_tanh(float x) {
  float t = __builtin_amdgcn_exp2f(x * 2.885390081777927f);  // 2*log2(e)
  float r = __builtin_amdgcn_rcpf(t + 1.0f);
  return __builtin_fmaf(-2.0f, r, 1.0f);
}

// Swap data between lane L and lane L^16 (wave32 half-swap) via ds_bpermute.
__device__ __forceinline__ float xor16(float v) {
  int lane = (int)(threadIdx.x & 31u);
  int i = __builtin_amdgcn_ds_bpermute((lane ^ 16) << 2, __builtin_bit_cast(int, v));
  return __builtin_bit_cast(float, i);
}

// One 16-batch Elman chain. ix[t][h] (with both biases folded in) is valid in
// lanes 0-15 (batch n = lane). A holds W_hh padded to 16x4. Hidden state lives
// as B-operand layout: VGPR0 = h0[n] (lanes 0-15), VGPR1 = h1[n]; lanes 16-31
// (K=2,3 rows) are zero and stay zero because the padded D rows are zero and
// tanh(0)==0.
__device__ __forceinline__ float rnn_chain(const float ix[4][2], v2f a, bool lo,
                                           float cw0, float cw1, float cb) {
  float h0 = lo ? fast_tanh(ix[0][0]) : 0.0f;  // h1 = tanh(ix_0) since h_0 = 0
  float h1 = lo ? fast_tanh(ix[0][1]) : 0.0f;
#pragma unroll
  for (int t = 1; t < 4; ++t) {
    v8f c;
    c[0] = lo ? ix[t][0] : 0.0f;  // C row 0 (lanes 16-31 = row 8 must be 0)
    c[1] = lo ? ix[t][1] : 0.0f;  // C row 1
    c[2] = 0.0f; c[3] = 0.0f; c[4] = 0.0f; c[5] = 0.0f; c[6] = 0.0f; c[7] = 0.0f;
    v2f b;
    b[0] = h0;  // K=0 row of B (lanes 16-31 already 0)
    b[1] = h1;  // K=1 row of B
    // D[m][n] = sum_k W_hh[m][k]*h_k[n] + ix_t[m][n]   (rows 2..15 stay 0)
    v8f d = __builtin_amdgcn_wmma_f32_16x16x4_f32(
        /*neg_a=*/false, a, /*neg_b=*/false, b,
        /*c_mod=*/(short)0, c, /*reuse_a=*/false, /*reuse_b=*/false);
    h0 = fast_tanh(d[0]);  // lanes 16-31: tanh(0) = 0, preserves padding
    h1 = fast_tanh(d[1]);
  }
  return __builtin_fmaf(cw0, h0, __builtin_fmaf(cw1, h1, cb));
}

__global__ __launch_bounds__(256) void rnn2cls_kernel(
    const float* __restrict__ X, const float* __restrict__ wih,
    const float* __restrict__ bih, const float* __restrict__ whh,
    const float* __restrict__ bhh, const float* __restrict__ clsw,
    const float* __restrict__ clsb, float* __restrict__ out, int B) {
  const int lane = (int)(threadIdx.x & 31u);
  const int wave = (int)(threadIdx.x >> 5u);
  const int base = (blockIdx.x * (int)(blockDim.x >> 5u) + wave) * 32;
  int idx = base + lane;
  idx = idx < B ? idx : (B - 1);  // clamp: EXEC must stay all-1s for WMMA

  // Coalesced: two b128 loads per lane, 32B-stride per lane -> full cachelines.
  const float4* xp = reinterpret_cast<const float4*>(X) + (size_t)idx * 2;
  const float4 xa = xp[0];  // t=0, t=1
  const float4 xb = xp[1];  // t=2, t=3

  // Uniform parameter loads (scalar path, L2-resident).
  const float w00 = wih[0], w01 = wih[1], w10 = wih[2], w11 = wih[3];
  const float b0 = bih[0] + bhh[0];
  const float b1 = bih[1] + bhh[1];
  const float cw0 = clsw[0], cw1 = clsw[1], cb = clsb[0];

  // A = W_hh padded to 16x4: lane m holds row m; VGPR0 = K=0 col, VGPR1 = K=1.
  v2f a;
  a[0] = lane == 0 ? whh[0] : (lane == 1 ? whh[2] : 0.0f);  // W_hh[m][0]
  a[1] = lane == 0 ? whh[1] : (lane == 1 ? whh[3] : 0.0f);  // W_hh[m][1]

  // ix[t][h] = W_ih[h][:] . x_t + b_ih[h] + b_hh[h]
  float ix[4][2];
  ix[0][0] = __builtin_fmaf(w00, xa.x, __builtin_fmaf(w01, xa.y, b0));
  ix[0][1] = __builtin_fmaf(w10, xa.x, __builtin_fmaf(w11, xa.y, b1));
  ix[1][0] = __builtin_fmaf(w00, xa.z, __builtin_fmaf(w01, xa.w, b0));
  ix[1][1] = __builtin_fmaf(w10, xa.z, __builtin_fmaf(w11, xa.w, b1));
  ix[2][0] = __builtin_fmaf(w00, xb.x, __builtin_fmaf(w01, xb.y, b0));
  ix[2][1] = __builtin_fmaf(w10, xb.x, __builtin_fmaf(w11, xb.y, b1));
  ix[3][0] = __builtin_fmaf(w00, xb.z, __builtin_fmaf(w01, xb.w, b0));
  ix[3][1] = __builtin_fmaf(w10, xb.z, __builtin_fmaf(w11, xb.w, b1));

  const bool lo = lane < 16;

  // Chain 1: batch elements base+0 .. base+15 (their ix is already in lanes 0-15).
  const float out1 = rnn_chain(ix, a, lo, cw0, cw1, cb);

  // Chain 2: batch elements base+16 .. base+31 — pull their ix down to lanes 0-15.
  float ix2[4][2];
#pragma unroll
  for (int t = 0; t < 4; ++t) {
    ix2[t][0] = xor16(ix[t][0]);
    ix2[t][1] = xor16(ix[t][1]);
  }
  const float out2 = rnn_chain(ix2, a, lo, cw0, cw1, cb);

  // Merge: lanes 16-31 take chain-2 results so the store is one coalesced b32.
  const float o2hi = xor16(out2);
  const float result = lo ? out1 : o2hi;
  if (base + lane < B) out[base + lane] = result;
}

extern "C" void kernel_launch(void* const* d_in, const int* in_sizes, int n_in,
                              void* d_out, int out_size, void* d_ws, size_t ws_size,
                              hipStream_t stream) {
  (void)n_in; (void)out_size; (void)d_ws; (void)ws_size;
  const float* X    = (const float*)d_in[0];
  const float* wih  = (const float*)d_in[1];
  const float* bih  = (const float*)d_in[2];
  const float* whh  = (const float*)d_in[3];
  const float* bhh  = (const float*)d_in[4];
  const float* clsw = (const float*)d_in[5];
  const float* clsb = (const float*)d_in[6];
  float* out = (float*)d_out;

  const int B = in_sizes[0] / 8;            // X is [B, 4, 2]
  const int blocks = (B + 255) / 256;       // 256 threads = 8 waves = 256 elements
  rnn2cls_kernel<<<blocks, 256, 0, stream>>>(X, wih, bih, whh, bhh, clsw, clsb,
                                             out, B);
}